// DoubleSAGEConv_13099650253558
// MI455X (gfx1250) — compile-verified
//
#include <hip/hip_runtime.h>
#include <math.h>

// ---------------------------------------------------------------------------
// DoubleSAGEConv (max-aggr) + BatchNorm + ReLU, twice.  F = 128.
// Pipeline per layer:
//   1) init agg = -inf
//   2) scatter-max over edges (float atomic max via int/uint trick: exact,
//      order independent -> deterministic)
//   3) tiled WMMA f32 GEMM: h = fix(agg)@W_l + x@W_r + b_l, with per-stripe
//      deterministic column sum/sumsq partials for BatchNorm
//   4) finalize BN stats -> per-feature scale/shift
//   5) elementwise BN + ReLU (in place)
// ---------------------------------------------------------------------------

#define F 128
#define TILE_M 16
#define LDS_STRIDE 132   // 128 + 4 pad -> conflict-free A-fragment reads

typedef __attribute__((ext_vector_type(2))) float v2f;
typedef __attribute__((ext_vector_type(8))) float v8f;

__device__ __forceinline__ void atomic_max_float(float* addr, float val) {
  // IEEE754 ordering trick: positives ordered as signed ints, negatives
  // reverse-ordered as unsigned ints. Init value is -inf (0xFF800000).
  if (val >= 0.0f) {
    atomicMax((int*)addr, __float_as_int(val));
  } else {
    atomicMin((unsigned int*)addr, __float_as_uint(val));
  }
}

__global__ void k_init_agg(unsigned int* __restrict__ agg, long long n) {
  long long i = (long long)blockIdx.x * blockDim.x + threadIdx.x;
  long long stride = (long long)gridDim.x * blockDim.x;
  for (; i < n; i += stride) agg[i] = 0xFF800000u;  // -inf
}

// One wave per edge; lane handles 4 consecutive features (float4 gather).
__global__ void k_scatter_max(const float* __restrict__ x,
                              const int* __restrict__ src,
                              const int* __restrict__ dst,
                              float* __restrict__ agg, int E) {
  long long t = (long long)blockIdx.x * blockDim.x + threadIdx.x;
  int e = (int)(t >> 5);
  if (e >= E) return;
  int lane = (int)(t & 31);
  int s = src[e];
  int d = dst[e];
  const float4 v = *(const float4*)(x + (long long)s * F + lane * 4);
  float* out = agg + (long long)d * F + lane * 4;
  atomic_max_float(out + 0, v.x);
  atomic_max_float(out + 1, v.y);
  atomic_max_float(out + 2, v.z);
  atomic_max_float(out + 3, v.w);
}

// Block = 256 threads = 8 waves; handles one 16-row stripe.
// Wave w computes the 16x16 output tile for columns [16w, 16w+16).
// K = 128 -> 32 k-steps, 2 WMMAs each (agg@Wl and x@Wr share accumulator).
__global__ void k_sage_gemm(const float* __restrict__ agg,
                            const float* __restrict__ xin,
                            const float* __restrict__ Wl,
                            const float* __restrict__ Wr,
                            const float* __restrict__ bias,
                            float* __restrict__ hout,
                            float* __restrict__ partials,  // [nstripes][256]
                            int N) {
  __shared__ float sA[TILE_M * LDS_STRIDE];
  __shared__ float sX[TILE_M * LDS_STRIDE];

  const int stripe = blockIdx.x;
  const int row0 = stripe * TILE_M;
  const int tid = threadIdx.x;

  // Stage 16x128 of agg (with -inf -> 0 fix) and x into LDS.
  for (int idx = tid; idx < TILE_M * (F / 4); idx += blockDim.x) {
    int r = idx >> 5;            // row within stripe (32 float4 per row)
    int c4 = (idx & 31) * 4;     // starting feature
    int row = row0 + r;
    float4 va = make_float4(0.f, 0.f, 0.f, 0.f);
    float4 vx = make_float4(0.f, 0.f, 0.f, 0.f);
    if (row < N) {
      va = *(const float4*)(agg + (long long)row * F + c4);
      vx = *(const float4*)(xin + (long long)row * F + c4);
    }
    // empty segments stay exactly -inf -> PyG semantics: replace with 0
    if (__float_as_uint(va.x) == 0xFF800000u) va.x = 0.f;
    if (__float_as_uint(va.y) == 0xFF800000u) va.y = 0.f;
    if (__float_as_uint(va.z) == 0xFF800000u) va.z = 0.f;
    if (__float_as_uint(va.w) == 0xFF800000u) va.w = 0.f;
    float* pa = sA + r * LDS_STRIDE + c4;
    float* px = sX + r * LDS_STRIDE + c4;
    pa[0] = va.x; pa[1] = va.y; pa[2] = va.z; pa[3] = va.w;
    px[0] = vx.x; px[1] = vx.y; px[2] = vx.z; px[3] = vx.w;
  }
  __syncthreads();

  const int lane = threadIdx.x & 31;
  const int wave = threadIdx.x >> 5;   // 0..7 -> column tile
  const int m    = lane & 15;          // row (A) / column (B,C)
  const int kh   = lane >> 4;          // K-half selector
  const int col  = wave * 16 + m;      // output feature column

  v8f c = {};
  for (int kk = 0; kk < F / 4; ++kk) {
    const int k0 = kk * 4 + kh * 2;
    v2f aA, aX, bl, br;
    // A 16x4 f32 layout: lane m (+16) holds A[m][2*kh + {0,1}]
    aA.x = sA[m * LDS_STRIDE + k0];
    aA.y = sA[m * LDS_STRIDE + k0 + 1];
    aX.x = sX[m * LDS_STRIDE + k0];
    aX.y = sX[m * LDS_STRIDE + k0 + 1];
    // B 4x16 f32 layout (mirror of A): lane holds B[2*kh + {0,1}][m]
    bl.x = Wl[(long long)k0 * F + col];
    bl.y = Wl[(long long)(k0 + 1) * F + col];
    br.x = Wr[(long long)k0 * F + col];
    br.y = Wr[(long long)(k0 + 1) * F + col];
    c = __builtin_amdgcn_wmma_f32_16x16x4_f32(false, aA, false, bl,
                                              (short)0, c, false, false);
    c = __builtin_amdgcn_wmma_f32_16x16x4_f32(false, aX, false, br,
                                              (short)0, c, false, false);
  }

  // C/D layout: VGPR r -> M = r + 8*kh, N = col. Add bias, store, and
  // accumulate per-column sum / sumsq for BatchNorm (deterministic partials).
  const float bv = bias[col];
  float s = 0.f, ss = 0.f;
#pragma unroll
  for (int r = 0; r < 8; ++r) {
    int row = row0 + r + kh * 8;
    float v = c[r] + bv;
    if (row < N) {
      hout[(long long)row * F + col] = v;
      s  += v;
      ss += v * v;
    }
  }
  // combine the two half-waves (same column, different rows)
  s  += __shfl_down(s, 16, 32);
  ss += __shfl_down(ss, 16, 32);
  if (lane < 16) {
    float* p = partials + (long long)stripe * 256;
    p[col]       = s;
    p[128 + col] = ss;
  }
}

__global__ void k_bn_finalize(const float* __restrict__ partials, int nstripes,
                              const float* __restrict__ gamma,
                              const float* __restrict__ beta,
                              float* __restrict__ scaleshift, float invN) {
  int f = threadIdx.x;  // 128 threads
  float s = 0.f, ss = 0.f;
  for (int b = 0; b < nstripes; ++b) {
    s  += partials[(long long)b * 256 + f];
    ss += partials[(long long)b * 256 + 128 + f];
  }
  float mu  = s * invN;
  float var = fmaxf(ss * invN - mu * mu, 0.0f);
  float sc  = gamma[f] * rsqrtf(var + 1e-5f);
  scaleshift[f]       = sc;
  scaleshift[128 + f] = beta[f] - mu * sc;
}

__global__ void k_bn_relu(float* __restrict__ h,
                          const float* __restrict__ scaleshift, long long n4) {
  long long i = (long long)blockIdx.x * blockDim.x + threadIdx.x;
  if (i >= n4) return;
  float4 v = ((float4*)h)[i];
  int f = (int)((i * 4) & (F - 1));
  float4 o;
  o.x = fmaxf(v.x * scaleshift[f + 0] + scaleshift[128 + f + 0], 0.f);
  o.y = fmaxf(v.y * scaleshift[f + 1] + scaleshift[128 + f + 1], 0.f);
  o.z = fmaxf(v.z * scaleshift[f + 2] + scaleshift[128 + f + 2], 0.f);
  o.w = fmaxf(v.w * scaleshift[f + 3] + scaleshift[128 + f + 3], 0.f);
  ((float4*)h)[i] = o;
}

extern "C" void kernel_launch(void* const* d_in, const int* in_sizes, int n_in,
                              void* d_out, int out_size, void* d_ws,
                              size_t ws_size, hipStream_t stream) {
  (void)n_in; (void)out_size; (void)ws_size;
  const float* x   = (const float*)d_in[0];
  const int*   ei  = (const int*)d_in[1];
  const float* W0l = (const float*)d_in[2];
  const float* b0l = (const float*)d_in[3];
  const float* W0r = (const float*)d_in[4];
  const float* g0  = (const float*)d_in[5];
  const float* be0 = (const float*)d_in[6];
  const float* W1l = (const float*)d_in[7];
  const float* b1l = (const float*)d_in[8];
  const float* W1r = (const float*)d_in[9];
  const float* g1  = (const float*)d_in[10];
  const float* be1 = (const float*)d_in[11];

  const int N = in_sizes[0] / F;
  const int E = in_sizes[1] / 2;
  const int* srcp = ei;
  const int* dstp = ei + E;
  const int nstripes = (N + TILE_M - 1) / TILE_M;
  const long long NF = (long long)N * F;

  float* agg        = (float*)d_ws;
  float* hbuf       = agg + NF;
  float* partials   = hbuf + NF;
  float* scaleshift = partials + (long long)nstripes * 256;
  float* outp       = (float*)d_out;

  dim3 blk(256);
  const unsigned scatterBlocks = (unsigned)(((long long)E * 32 + 255) / 256);
  const long long n4 = NF / 4;
  const unsigned ewBlocks = (unsigned)((n4 + 255) / 256);
  const float invN = 1.0f / (float)N;

  // ---- layer 0 ----
  k_init_agg<<<2048, blk, 0, stream>>>((unsigned*)agg, NF);
  k_scatter_max<<<scatterBlocks, blk, 0, stream>>>(x, srcp, dstp, agg, E);
  k_sage_gemm<<<nstripes, blk, 0, stream>>>(agg, x, W0l, W0r, b0l, hbuf,
                                            partials, N);
  k_bn_finalize<<<1, F, 0, stream>>>(partials, nstripes, g0, be0, scaleshift,
                                     invN);
  k_bn_relu<<<ewBlocks, blk, 0, stream>>>(hbuf, scaleshift, n4);

  // ---- layer 1 ----
  k_init_agg<<<2048, blk, 0, stream>>>((unsigned*)agg, NF);
  k_scatter_max<<<scatterBlocks, blk, 0, stream>>>(hbuf, srcp, dstp, agg, E);
  k_sage_gemm<<<nstripes, blk, 0, stream>>>(agg, hbuf, W1l, W1r, b1l, outp,
                                            partials, N);
  k_bn_finalize<<<1, F, 0, stream>>>(partials, nstripes, g1, be1, scaleshift,
                                     invN);
  k_bn_relu<<<ewBlocks, blk, 0, stream>>>(outp, scaleshift, n4);
}